// FNO3DForecaster_81389630259612
// MI455X (gfx1250) — compile-verified
//
#include <hip/hip_runtime.h>

#define DEV __device__ __forceinline__

typedef __attribute__((ext_vector_type(2))) float v2f;
typedef __attribute__((ext_vector_type(8))) float v8f;
typedef __attribute__((ext_vector_type(4))) unsigned int u32x4;
typedef __attribute__((ext_vector_type(8))) unsigned int u32x8;

constexpr int Bb = 8, Hh = 64, Ww = 64, Tt = 32;
constexpr int WID = 32, NL = 4;
constexpr int NPT   = Bb * Hh * Ww * Tt;   // 1,048,576 points
constexpr int NLINE = Bb * Hh * Ww;        // 32768 (b,h,w) lines

// ---- workspace float offsets ----
constexpr size_t O_FT  = 0;                     // Ft  [16][32]   fwd time DFT (rows: 8 cos, 8 -sin)
constexpr size_t O_FW  = 512;                   // Fw  [32][128]  fwd 64-pt DFT (shared H/W), K=(pos,ri)
constexpr size_t O_EH  = 4608;                  // Eh  [128][32]  inv 64-pt DFT (shared H/W), M=(pos,ri)
constexpr size_t O_ET  = 8704;                  // Et  [32][16]   inv time (real combine)
constexpr size_t O_X   = 16384;
constexpr size_t SZ_X  = (size_t)NPT * WID;                 // 33,554,432
constexpr size_t O_Y   = O_X + SZ_X;
constexpr size_t O_G1  = O_Y + SZ_X;                        // [line(b,h,w)][ktri16][c32]   (also Z1 alias)
constexpr size_t SZ_G1 = (size_t)NLINE * 16 * WID;          // 16,777,216
constexpr size_t O_G2  = O_G1 + SZ_G1;                      // [b*h][kwri32][kt8*c32]       (also Z2 alias)
constexpr size_t SZ_G2 = (size_t)Bb * Hh * 32 * 256;        // 4,194,304
constexpr size_t O_G3  = O_G2 + SZ_G2;                      // [b][khri32][kw16][kt8*c32]
constexpr size_t SZ_G3 = (size_t)Bb * 32 * 16 * 256;        // 1,048,576
constexpr size_t O_G4  = O_G3 + SZ_G3;                      // same shape as G3, channels mixed

// ---------------- TDM: DMA a 1-D run of fp32 from global into LDS ----------------
// Builds the D# (group0 128b + group1 256b) in SGPRs and issues TENSOR_LOAD_TO_LDS.
// Tracked by TENSORcnt; EXEC is ignored, so callers gate by control flow (wave 0 only).
DEV void tdm_load_1d(void* ldsDst, const void* gsrc, unsigned nElem) {
  unsigned ldsOff = (unsigned)(unsigned long long)ldsDst;     // low 32b of LDS aperture addr
  unsigned long long ga = (unsigned long long)gsrc;
  u32x4 g0;
  g0[0] = 1u;                                   // count=1, user descriptor, no gather
  g0[1] = ldsOff;                               // lds_addr
  g0[2] = (unsigned)ga;                         // global_addr[31:0]
  g0[3] = (unsigned)(ga >> 32) | (2u << 30);    // global_addr[56:32] | type=2 (image)
  u32x8 g1;
  g1[0] = 0x00020000u;                          // workgroup_mask=0, data_size=2 (4 bytes)
  g1[1] = (nElem & 0xFFFFu) << 16;              // tensor_dim0[15:0]
  g1[2] = (nElem >> 16) | (1u << 16);           // tensor_dim0[31:16] | tensor_dim1=1
  g1[3] = (nElem & 0xFFFFu) << 16;              // tile_dim0 = nElem (<=65535)
  g1[4] = 0u;                                   // tile_dim1=0 (unused), tile_dim2=0
  g1[5] = nElem;                                // tensor_dim0_stride[31:0]
  g1[6] = 0u;                                   // stride hi / dim1_stride lo
  g1[7] = 0u;
  asm volatile("tensor_load_to_lds %0, %1" :: "s"(g0), "s"(g1) : "memory");
}
DEV void tdm_wait_all() { __builtin_amdgcn_s_wait_tensorcnt(0); }

// ---------------- WMMA helpers (fp32 16x16x4, per ISA VGPR layouts) ----------------
DEV v8f wmma4(v2f a, v2f b, v8f c) {
  return __builtin_amdgcn_wmma_f32_16x16x4_f32(false, a, false, b, (short)0, c, false, false);
}
// A 16x4 (MxK): lane L holds row L%16; VGPR0/1 = K = k0+2*(L/16), +1
DEV v2f ldA(const float* src, int ld, int k0, int lane) {
  int row = lane & 15, kb = k0 + ((lane >> 4) << 1);
  v2f a; a.x = src[row * ld + kb]; a.y = src[row * ld + kb + 1]; return a;
}
// B 4x16 (KxN), row-major [K][N]: lane L holds col n0+L%16; VGPR0/1 = rows kb, kb+1
DEV v2f ldB(const float* src, int ld, int k0, int n0, int lane) {
  int col = n0 + (lane & 15), kb = k0 + ((lane >> 4) << 1);
  v2f b; b.x = src[(size_t)kb * ld + col]; b.y = src[(size_t)(kb + 1) * ld + col]; return b;
}
// C/D 16x16: VGPR r -> row r + 8*(L/16), col L%16
DEV void stC(float* dst, int ld, v8f c, int lane) {
  int col = lane & 15, rb = (lane >> 4) << 3;
#pragma unroll
  for (int r = 0; r < 8; ++r) dst[(size_t)(rb + r) * ld + col] = c[r];
}
DEV float gelu(float x) { return 0.5f * x * (1.0f + erff(x * 0.70710678118654752f)); }

// ---------------- DFT matrix construction (once per launch) ----------------
__global__ void k_init(float* __restrict__ M) {
  const float P2 = 6.28318530717958647692f;
  for (int e = threadIdx.x + blockIdx.x * blockDim.x; e < 9216; e += blockDim.x * gridDim.x) {
    float v = 0.f;
    if (e < 512) {                               // Ft[16][32]: rfft real rows then imag rows
      int r = e >> 5, t = e & 31;
      float a = P2 * (float)(((r & 7) * t) & 31) / 32.f;
      v = (r < 8) ? __cosf(a) : -__sinf(a);
    } else if (e < 4608) {                       // Fw[32][128]: out=(kwri), K=(pos,ri_in)
      int i = e - 512; int m = i >> 7, k = i & 127;
      int f = m & 15; int fr = (f < 8) ? f : f + 48;
      int pos = k >> 1, ri = k & 1;
      float a = P2 * (float)((fr * pos) & 63) / 64.f;
      v = (m < 16) ? (ri ? __sinf(a) : __cosf(a))     // out_r = sum xr*cos + xi*sin
                   : (ri ? __cosf(a) : -__sinf(a));   // out_i = sum xi*cos - xr*sin
    } else if (e < 8704) {                       // Eh[128][32]: M=(pos,ri_out), K=(f,ri_in), 1/64
      int i = e - 4608; int m = i >> 5, k = i & 31;
      int riO = m >> 6, pos = m & 63;
      int riI = k >> 4, f = k & 15; int fr = (f < 8) ? f : f + 48;
      float a = P2 * (float)((fr * pos) & 63) / 64.f; const float s = 1.f / 64.f;
      v = (riO == 0) ? (riI ? -__sinf(a) * s : __cosf(a) * s)
                     : (riI ?  __cosf(a) * s : __sinf(a) * s);
    } else {                                     // Et[32][16]: irfft combine, 1/32, Im(bin0) dropped
      int i = e - 8704; int t = i >> 4, k = i & 15;
      int ri = k >> 3, kt = k & 7;
      float a = P2 * (float)((kt * t) & 31) / 32.f;
      v = (ri == 0) ? ((kt == 0) ? (1.f / 32.f) : (2.f / 32.f) * __cosf(a))
                    : ((kt == 0) ? 0.f : (-2.f / 32.f) * __sinf(a));
    }
    M[e] = v;
  }
}

// ---------------- lift: x = P @ [x_t; gy; gx; gt] + b  -> X[point][32] ----------------
__global__ void k_lift(const float* __restrict__ xt, const float* __restrict__ pw,
                       const float* __restrict__ pb, float* __restrict__ X) {
  int p = blockIdx.x * blockDim.x + threadIdx.x;
  if (p >= NPT) return;
  int t = p & 31, w = (p >> 5) & 63, h = (p >> 11) & 63, b = p >> 17;
  float v[6];
  v[0] = xt[(((size_t)b * 3 + 0) * 64 + h) * 64 + w];
  v[1] = xt[(((size_t)b * 3 + 1) * 64 + h) * 64 + w];
  v[2] = xt[(((size_t)b * 3 + 2) * 64 + h) * 64 + w];
  v[3] = (float)h * (1.f / 63.f);
  v[4] = (float)w * (1.f / 63.f);
  v[5] = (float)t * (1.f / 31.f);
  float* o = X + (size_t)p * WID;
#pragma unroll
  for (int c = 0; c < WID; ++c) {
    float s = pb[c];
#pragma unroll
    for (int i = 0; i < 6; ++i) s = fmaf(pw[c * 6 + i], v[i], s);
    o[c] = s;
  }
}

// ---------------- forward time DFT: G1[line][ktri16][c32] = Ft(16x32) @ X[line](32x32) ----------------
__global__ void k_dft_t(const float* __restrict__ X, const float* __restrict__ M,
                        float* __restrict__ G1) {
  __shared__ float sFt[512];
  if (threadIdx.x < 32) tdm_load_1d(sFt, M + O_FT, 512);
  tdm_wait_all();
  __syncthreads();
  int lane = threadIdx.x & 31;
  int work = (blockIdx.x * blockDim.x + threadIdx.x) >> 5;  // NLINE*2
  int nt = work & 1, line = work >> 1;
  if (line >= NLINE) return;
  const float* src = X + (size_t)line * Tt * WID;
  float* dst = G1 + (size_t)line * 16 * WID + nt * 16;
  v8f acc = {0.f,0.f,0.f,0.f,0.f,0.f,0.f,0.f};
#pragma unroll
  for (int k0 = 0; k0 < 32; k0 += 4)
    acc = wmma4(ldA(sFt, 32, k0, lane), ldB(src, 32, k0, nt * 16, lane), acc);
  stC(dst, WID, acc, lane);
}

// ---------------- forward W DFT (complex, K=(w,ri)=128): G2[bh][kwri32][kt*c 256] ----------------
__global__ void k_dft_w(const float* __restrict__ G1, const float* __restrict__ M,
                        float* __restrict__ G2) {
  __shared__ float sFw[4096];
  if (threadIdx.x < 32) tdm_load_1d(sFw, M + O_FW, 4096);
  tdm_wait_all();
  __syncthreads();
  int lane = threadIdx.x & 31;
  int work = (blockIdx.x * blockDim.x + threadIdx.x) >> 5;  // 512*2*16
  int nt = work & 15, mt = (work >> 4) & 1, line = work >> 5;  // line=(b,h)
  if (line >= Bb * Hh) return;
  const float* Fw  = sFw + mt * 16 * 128;
  const float* src = G1 + (size_t)line * Ww * 16 * WID;   // w stride 512 floats
  float* dst = G2 + (size_t)line * 32 * 256 + (size_t)mt * 16 * 256 + nt * 16;
  int col = nt * 16 + (lane & 15);                        // (kt*32+c)
  v8f acc = {0.f,0.f,0.f,0.f,0.f,0.f,0.f,0.f};
  for (int k0 = 0; k0 < 128; k0 += 4) {
    v2f a = ldA(Fw, 128, k0, lane);
    int kb = k0 + ((lane >> 4) << 1);                     // even: ri=0
    const float* bp = src + (size_t)(kb >> 1) * 512 + col;
    v2f b; b.x = bp[0]; b.y = bp[256];                    // ri stride = 8*32
    acc = wmma4(a, b, acc);
  }
  stC(dst, 256, acc, lane);
}

// ---------------- forward H DFT: G3[b][khri32][kw16][kt*c 256] ----------------
__global__ void k_dft_h(const float* __restrict__ G2, const float* __restrict__ M,
                        float* __restrict__ G3) {
  __shared__ float sFw[4096];
  if (threadIdx.x < 32) tdm_load_1d(sFw, M + O_FW, 4096);
  tdm_wait_all();
  __syncthreads();
  int lane = threadIdx.x & 31;
  int work = (blockIdx.x * blockDim.x + threadIdx.x) >> 5;  // 8*2*256
  int nt = work & 255, mt = (work >> 8) & 1, b = work >> 9;
  if (b >= Bb) return;
  const float* Fh  = sFw + mt * 16 * 128;
  const float* src = G2 + (size_t)b * Hh * 32 * 256;        // h stride 8192 floats
  float* dst = G3 + (size_t)b * 32 * 4096 + (size_t)mt * 16 * 4096 + nt * 16;
  int n = nt * 16 + (lane & 15);                            // (kw,kt,c) in [0,4096)
  int kw = n >> 8, rem = n & 255;
  v8f acc = {0.f,0.f,0.f,0.f,0.f,0.f,0.f,0.f};
  for (int k0 = 0; k0 < 128; k0 += 4) {
    v2f a = ldA(Fh, 128, k0, lane);
    int kb = k0 + ((lane >> 4) << 1);
    const float* bp = src + (size_t)(kb >> 1) * 8192 + (size_t)kw * 256 + rem;
    v2f b; b.x = bp[0]; b.y = bp[4096];                     // ri stride = 16*256
    acc = wmma4(a, b, acc);
  }
  stC(dst, 4096, acc, lane);
}

// ---------------- per-mode complex channel mix (corner weights) ----------------
__global__ void k_mix(const float* __restrict__ G3, const float* __restrict__ wr,
                      const float* __restrict__ wi, float* __restrict__ G4, int layer) {
  __shared__ float sWr[1024], sWi[1024], sXr[256], sXi[256];
  int m = blockIdx.x;  // (kh*16+kw)*8+kt
  int kh = m >> 7, kw = (m >> 3) & 15, kt = m & 7;
  int corner = (kh >= 8 ? 1 : 0) + (kw >= 8 ? 2 : 0);
  size_t wb = (size_t)(layer * 4 + corner) * 1024 * 512
            + (size_t)(kh & 7) * 64 + (size_t)(kw & 7) * 8 + kt;
  int tid = threadIdx.x;
  for (int e = tid; e < 1024; e += 256) {
    size_t idx = wb + (size_t)e * 512;   // e = i*32+o
    sWr[e] = wr[idx]; sWi[e] = wi[idx];
  }
  {
    int b = tid >> 5, i = tid & 31;
    size_t xb = (size_t)b * 32 * 4096 + (size_t)kh * 4096 + (size_t)kw * 256 + kt * 32 + i;
    sXr[tid] = G3[xb];
    sXi[tid] = G3[xb + (size_t)16 * 4096];
  }
  __syncthreads();
  int b = tid >> 5, o = tid & 31;
  float yr = 0.f, yi = 0.f;
#pragma unroll
  for (int i = 0; i < 32; ++i) {
    float xr = sXr[b * 32 + i], xi = sXi[b * 32 + i];
    float r = sWr[i * 32 + o], im = sWi[i * 32 + o];
    yr = fmaf(xr, r, yr); yr = fmaf(-xi, im, yr);
    yi = fmaf(xr, im, yi); yi = fmaf(xi, r, yi);
  }
  size_t ob = (size_t)b * 32 * 4096 + (size_t)kh * 4096 + (size_t)kw * 256 + kt * 32 + o;
  G4[ob] = yr; G4[ob + (size_t)16 * 4096] = yi;
}

// ---------------- inverse H: Z2[b][(ri*64+h)128][kw16][kt*c 256] = Eh(128x32) @ G4[b] ----------------
__global__ void k_idft_h(const float* __restrict__ G4, const float* __restrict__ M,
                         float* __restrict__ Z2) {
  __shared__ float sEh[4096];
  if (threadIdx.x < 32) tdm_load_1d(sEh, M + O_EH, 4096);
  tdm_wait_all();
  __syncthreads();
  int lane = threadIdx.x & 31;
  int work = (blockIdx.x * blockDim.x + threadIdx.x) >> 5;  // 8*8*256
  int nt = work & 255, mt = (work >> 8) & 7, b = work >> 11;
  if (b >= Bb) return;
  const float* Eh  = sEh + mt * 16 * 32;
  const float* src = G4 + (size_t)b * 32 * 4096;            // plain [32][4096]
  float* dst = Z2 + (size_t)b * 128 * 4096 + (size_t)mt * 16 * 4096 + nt * 16;
  v8f acc = {0.f,0.f,0.f,0.f,0.f,0.f,0.f,0.f};
#pragma unroll
  for (int k0 = 0; k0 < 32; k0 += 4)
    acc = wmma4(ldA(Eh, 32, k0, lane), ldB(src, 4096, k0, nt * 16, lane), acc);
  stC(dst, 4096, acc, lane);
}

// ---------------- inverse W: Z1[bh][(ri*64+w)128][kt*c 256], K=(ri_in*16+kw)=32 ----------------
__global__ void k_idft_w(const float* __restrict__ Z2, const float* __restrict__ M,
                         float* __restrict__ Z1) {
  __shared__ float sEh[4096];
  if (threadIdx.x < 32) tdm_load_1d(sEh, M + O_EH, 4096);
  tdm_wait_all();
  __syncthreads();
  int lane = threadIdx.x & 31;
  int work = (blockIdx.x * blockDim.x + threadIdx.x) >> 5;  // 512*8*16
  int nt = work & 15, mt = (work >> 4) & 7, line = work >> 7;  // line=b*64+h
  if (line >= Bb * Hh) return;
  int b = line >> 6, h = line & 63;
  const float* Ew  = sEh + mt * 16 * 32;
  const float* src = Z2 + (size_t)b * 128 * 4096 + (size_t)h * 4096;
  float* dst = Z1 + (size_t)line * 128 * 256 + (size_t)mt * 16 * 256 + nt * 16;
  int n = nt * 16 + (lane & 15);
  v8f acc = {0.f,0.f,0.f,0.f,0.f,0.f,0.f,0.f};
#pragma unroll
  for (int k0 = 0; k0 < 32; k0 += 4) {
    v2f a = ldA(Ew, 32, k0, lane);
    int kb = k0 + ((lane >> 4) << 1);
    const float* bp = src + (size_t)(kb >> 4) * 64 * 4096 + (size_t)(kb & 15) * 256 + n;
    v2f bv; bv.x = bp[0]; bv.y = bp[256];
    acc = wmma4(a, bv, acc);
  }
  stC(dst, 256, acc, lane);
}

// ---------------- fused: inverse-T (Et 32x16) + pointwise conv + bias + GELU -> Y ----------------
__global__ void k_fused(const float* __restrict__ Z1, const float* __restrict__ X,
                        const float* __restrict__ M, const float* __restrict__ pw,
                        const float* __restrict__ pb, float* __restrict__ Y) {
  __shared__ float sEt[512], sPw[1024];
  if (threadIdx.x < 32) {
    tdm_load_1d(sEt, M + O_ET, 512);
    tdm_load_1d(sPw, pw, 1024);
  }
  tdm_wait_all();
  __syncthreads();
  int lane = threadIdx.x & 31;
  int work = (blockIdx.x * blockDim.x + threadIdx.x) >> 5;  // NLINE*4
  int nt = work & 1, mt = (work >> 1) & 1, line = work >> 2;
  if (line >= NLINE) return;
  int w = line & 63, bh = line >> 6;
  const float* Et = sEt + mt * 16 * 16;
  const float* z  = Z1 + (size_t)bh * 128 * 256;
  const float* xs = X + (size_t)line * 1024 + mt * 16 * 32;
  int col = nt * 16 + (lane & 15);
  v8f acc = {0.f,0.f,0.f,0.f,0.f,0.f,0.f,0.f};
#pragma unroll
  for (int k0 = 0; k0 < 16; k0 += 4) {        // spectral: K = (ri*8+kt) = 16
    v2f a = ldA(Et, 16, k0, lane);
    int kb = k0 + ((lane >> 4) << 1);
    const float* bp = z + ((size_t)(kb >> 3) * 64 + w) * 256 + (size_t)(kb & 7) * 32 + col;
    v2f bv; bv.x = bp[0]; bv.y = bp[32];
    acc = wmma4(a, bv, acc);
  }
#pragma unroll
  for (int k0 = 0; k0 < 32; k0 += 4) {        // pointwise: K = i = 32
    v2f a = ldA(xs, 32, k0, lane);
    int kb = k0 + ((lane >> 4) << 1);
    v2f bv; bv.x = sPw[col * 32 + kb]; bv.y = sPw[col * 32 + kb + 1];
    acc = wmma4(a, bv, acc);
  }
  float bias = pb[col];
  float* dst = Y + (size_t)line * 1024;
  int rb = (lane >> 4) << 3;
#pragma unroll
  for (int r = 0; r < 8; ++r) {
    int t = mt * 16 + rb + r;
    dst[t * 32 + col] = gelu(acc[r] + bias);
  }
}

// ---------------- head: q1 (WMMA) + GELU -> LDS -> q2 + transposed store ----------------
__global__ void k_head(const float* __restrict__ X, const float* __restrict__ q1w,
                       const float* __restrict__ q1b, const float* __restrict__ q2w,
                       const float* __restrict__ q2b, float* __restrict__ out) {
  __shared__ float sQ1[2048];
  __shared__ float h1[8][32 * 64];
  if (threadIdx.x < 32) tdm_load_1d(sQ1, q1w, 2048);
  tdm_wait_all();
  __syncthreads();
  int lane = threadIdx.x & 31, wave = threadIdx.x >> 5;
  int line = blockIdx.x * 8 + wave;
  if (line >= NLINE) return;
  int w = line & 63, h = (line >> 6) & 63, b = line >> 12;
  const float* xs = X + (size_t)line * 1024;
  float* hl = h1[wave];
  for (int mt = 0; mt < 2; ++mt)
    for (int nt = 0; nt < 4; ++nt) {
      int col = nt * 16 + (lane & 15);
      v8f acc = {0.f,0.f,0.f,0.f,0.f,0.f,0.f,0.f};
#pragma unroll
      for (int k0 = 0; k0 < 32; k0 += 4) {
        v2f a = ldA(xs + mt * 16 * 32, 32, k0, lane);
        int kb = k0 + ((lane >> 4) << 1);
        v2f bv; bv.x = sQ1[col * 32 + kb]; bv.y = sQ1[col * 32 + kb + 1];
        acc = wmma4(a, bv, acc);
      }
      float bias = q1b[col];
      int rb = (lane >> 4) << 3;
#pragma unroll
      for (int r = 0; r < 8; ++r) {
        int t = mt * 16 + rb + r;
        hl[t * 64 + col] = gelu(acc[r] + bias);
      }
    }
  __syncthreads();
  int t = lane;
  const float* hv = hl + t * 64;
#pragma unroll
  for (int c = 0; c < 3; ++c) {
    float s = q2b[c];
#pragma unroll
    for (int j = 0; j < 64; ++j) s = fmaf(q2w[c * 64 + j], hv[j], s);
    out[(((size_t)b * 32 + t) * 3 + c) * 4096 + h * 64 + w] = s;
  }
}

extern "C" void kernel_launch(void* const* d_in, const int* in_sizes, int n_in,
                              void* d_out, int out_size, void* d_ws, size_t ws_size,
                              hipStream_t stream) {
  (void)in_sizes; (void)n_in; (void)out_size; (void)ws_size;
  const float* xt  = (const float*)d_in[0];
  const float* p_w = (const float*)d_in[1];
  const float* p_b = (const float*)d_in[2];
  const float* swr = (const float*)d_in[3];
  const float* swi = (const float*)d_in[4];
  const float* pww = (const float*)d_in[5];
  const float* pwb = (const float*)d_in[6];
  const float* q1w = (const float*)d_in[7];
  const float* q1b = (const float*)d_in[8];
  const float* q2w = (const float*)d_in[9];
  const float* q2b = (const float*)d_in[10];
  float* ws = (float*)d_ws;
  float* M  = ws;            // DFT matrices
  float* Xa = ws + O_X;
  float* Xb = ws + O_Y;
  float* G1 = ws + O_G1;     // also Z1
  float* G2 = ws + O_G2;     // also Z2
  float* G3 = ws + O_G3;
  float* G4 = ws + O_G4;
  float* out = (float*)d_out;

  k_init<<<36, 256, 0, stream>>>(M);
  k_lift<<<NPT / 256, 256, 0, stream>>>(xt, p_w, p_b, Xa);

  float* cur = Xa; float* nxt = Xb;
  for (int l = 0; l < NL; ++l) {
    k_dft_t <<<(NLINE * 2) / 8,      256, 0, stream>>>(cur, M, G1);
    k_dft_w <<<(512 * 2 * 16) / 8,   256, 0, stream>>>(G1, M, G2);
    k_dft_h <<<(8 * 2 * 256) / 8,    256, 0, stream>>>(G2, M, G3);
    k_mix   <<<2048,                 256, 0, stream>>>(G3, swr, swi, G4, l);
    k_idft_h<<<(8 * 8 * 256) / 8,    256, 0, stream>>>(G4, M, G2);   // Z2 aliases G2
    k_idft_w<<<(512 * 8 * 16) / 8,   256, 0, stream>>>(G2, M, G1);   // Z1 aliases G1
    k_fused <<<(NLINE * 4) / 8,      256, 0, stream>>>(G1, cur, M, pww + l * 1024, pwb + l * 32, nxt);
    float* tmp = cur; cur = nxt; nxt = tmp;
  }
  k_head<<<NLINE / 8, 256, 0, stream>>>(cur, q1w, q1b, q2w, q2b, out);
}